// HoltWinters_6416681141104
// MI455X (gfx1250) — compile-verified
//
#include <hip/hip_runtime.h>
#include <stdint.h>

// ---------------- problem constants ----------------
#define T_LEN    512
#define SEASLEN  12
#define TILE     32          // timesteps per wave-tile
#define WPB      4           // waves per block
#define BLOCK_T  (WPB * 32)  // 128 threads
#define LSTRIDE  33          // 32 + 1 pad float (bank-conflict-free, matches TDM pad)

#define AS3 __attribute__((address_space(3)))
#define AS1 __attribute__((address_space(1)))

#if defined(__has_builtin)
#define HB(x) __has_builtin(x)
#else
#define HB(x) 0
#endif

#if HB(__builtin_amdgcn_tensor_load_to_lds)
#define HAVE_TDM 1
#else
#define HAVE_TDM 0
#endif

__device__ __forceinline__ void cfence() { asm volatile("" ::: "memory"); }

__device__ __forceinline__ void wait_async0() {
#if HB(__builtin_amdgcn_s_wait_asynccnt)
    __builtin_amdgcn_s_wait_asynccnt(0);
#else
    asm volatile("s_wait_asynccnt 0x0" ::: "memory");
#endif
}

__device__ __forceinline__ void wait_ds0() {
#if HB(__builtin_amdgcn_s_wait_dscnt)
    __builtin_amdgcn_s_wait_dscnt(0);
#else
    asm volatile("s_wait_dscnt 0x0" ::: "memory");
#endif
}

__device__ __forceinline__ void wait_tensor0() {
#if HB(__builtin_amdgcn_s_wait_tensorcnt)
    __builtin_amdgcn_s_wait_tensorcnt(0);
#else
    asm volatile("s_wait_tensorcnt 0x0" ::: "memory");
#endif
}

__device__ __forceinline__ void wait_tensor1() {
#if HB(__builtin_amdgcn_s_wait_tensorcnt)
    __builtin_amdgcn_s_wait_tensorcnt(1);
#else
    asm volatile("s_wait_tensorcnt 0x1" ::: "memory");
#endif
}

__device__ __forceinline__ unsigned lds_off_u32(const float* p) {
    return (unsigned)(unsigned long)(AS3 const float*)p;
}

// Async DMA: LDS -> global, 4 bytes per lane, ASYNCcnt-tracked.
__device__ __forceinline__ void async_store_f32(float* g, float* l) {
#if HB(__builtin_amdgcn_global_store_async_from_lds_b32)
    __builtin_amdgcn_global_store_async_from_lds_b32(
        (AS1 int*)g, (AS3 int*)l, 0, 0);
#else
    unsigned loff = lds_off_u32(l);
    unsigned long long ga = (unsigned long long)(uintptr_t)g;
    asm volatile("global_store_async_from_lds_b32 %0, %1, off"
                 :: "v"(ga), "v"(loff) : "memory");
#endif
}

#if !HAVE_TDM
// Fallback: async DMA global -> LDS, 4 bytes per lane.
__device__ __forceinline__ void async_load_f32(const float* g, float* l) {
#if HB(__builtin_amdgcn_global_load_async_to_lds_b32)
    __builtin_amdgcn_global_load_async_to_lds_b32(
        (AS1 int*)g, (AS3 int*)l, 0, 0);
#else
    unsigned loff = lds_off_u32(l);
    unsigned long long ga = (unsigned long long)(uintptr_t)g;
    asm volatile("global_load_async_to_lds_b32 %0, %1, off"
                 :: "v"(loff), "v"(ga) : "memory");
#endif
}
#endif

#if HAVE_TDM
typedef unsigned int u32x4 __attribute__((ext_vector_type(4)));
typedef int          i32x4 __attribute__((ext_vector_type(4)));
typedef int          i32x8 __attribute__((ext_vector_type(8)));

// One TDM op moves a 32(rows=series) x 32(cols=time) f32 tile from global
// (row pitch 512 floats) into LDS, inserting 1 DWORD of pad after every
// 32 DWORDs -> LDS row stride = 33 floats (bank-conflict-free).
__device__ __forceinline__ void tdm_load_tile32x32(const float* g, unsigned lds_byte) {
    unsigned long long ga = (unsigned long long)(uintptr_t)g;
    u32x4 g0 = { 1u,                                   // count=1 (valid), user mode
                 lds_byte,                             // lds_addr (bytes)
                 (unsigned)ga,                         // global_addr[31:0]
                 (unsigned)((ga >> 32) & 0x01FFFFFFu)  // global_addr[56:32]
                   | 0x80000000u };                    // type=2 ("image")
    i32x8 g1 = { (int)((2u << 16)   // data_size = 4 bytes
                     | (1u << 20)   // pad_enable
                     | (4u << 22)), // pad_interval: 32 DWORDs  (pad_amount=0 -> 1 DWORD)
                 (int)0xFFFF0000u,  // atomic_barrier_addr=0 | tensor_dim0.lo16 (dim0=0x7FFFFFFF)
                 (int)0xFFFF7FFFu,  // tensor_dim0.hi16       | tensor_dim1.lo16 (dim1=0x7FFFFFFF)
                 (int)((32u << 16) | 0x7FFFu), // tensor_dim1.hi16 | tile_dim0=32
                 (int)32,           // tile_dim1=32 | tile_dim2=0
                 (int)T_LEN,        // tensor_dim0_stride.lo32 = 512 elements
                 0, 0 };            // stride hi / dim1_stride = 0
    i32x4 z4 = { 0, 0, 0, 0 };
    i32x8 z8 = { 0, 0, 0, 0, 0, 0, 0, 0 };
    // 6-arg (clang-23 / therock) form: (g0, g1, g2, g3, extra, cpol)
    __builtin_amdgcn_tensor_load_to_lds(g0, g1, z4, z4, z8, 0);
}
#endif

// One thread owns one series. Each wave owns 32 adjacent series; input tiles
// are DMA'd by the Tensor Data Mover into a double-buffered padded LDS tile
// (prefetch pipelined with s_wait_tensorcnt), results staged transposed in LDS
// and drained with async LDS->global stores. All HBM traffic is 128B-coalesced.
__global__ __launch_bounds__(BLOCK_T)
void holtwinters_scan(const float* __restrict__ series,
                      const int*   __restrict__ shifts,
                      const float* __restrict__ alpha_p,
                      const float* __restrict__ beta_p,
                      const float* __restrict__ season_p,
                      const float* __restrict__ trend_p,
                      float*       __restrict__ out)
{
    __shared__ float tile_in [2][WPB][TILE * LSTRIDE]; // row-major: [series][33]
    __shared__ float tile_out[WPB][TILE * LSTRIDE];    // col-major: [time][33]
    __shared__ float season_s[16];

    const int tid  = threadIdx.x;
    const int lane = tid & 31;
    const int wave = tid >> 5;

    if (tid < SEASLEN) season_s[tid] = season_p[tid];
    __syncthreads();

    const float alpha      = alpha_p[0];
    const float beta       = beta_p[0];
    const float init_trend = trend_p[0];

    const int b  = blockIdx.x * BLOCK_T + tid;        // series owned by this lane
    const int b0 = blockIdx.x * BLOCK_T + wave * 32;  // first series of this wave

    // season index for time t is (t - shift) mod 12
    const int shift = shifts[b];
    int j = (SEASLEN - shift) % SEASLEN;

    float trend = 0.f;
    float prev  = 0.f;   // smooth + trend carried between steps

    const float* gser = series + (size_t)b0 * T_LEN;
    float* tout = &tile_out[wave][0];

#if HAVE_TDM
    tdm_load_tile32x32(gser, lds_off_u32(&tile_in[0][wave][0]));   // prologue: tile 0
#endif

    for (int t0 = 0; t0 < T_LEN; t0 += TILE) {
        const int buf = (t0 >> 5) & 1;
        float* tin = &tile_in[buf][wave][0];

#if HAVE_TDM
        if (t0 + TILE < T_LEN) {   // prefetch next tile into other buffer
            tdm_load_tile32x32(gser + t0 + TILE,
                               lds_off_u32(&tile_in[buf ^ 1][wave][0]));
            wait_tensor1();        // current tile's TDM done (newest may fly)
        } else {
            wait_tensor0();
        }
#else
        {   // fallback: 32 async b32 DMAs, instr i = series row i, lane = time col
            const float* gbase = gser + t0 + lane;
            #pragma unroll
            for (int i = 0; i < TILE; ++i)
                async_load_f32(gbase + (size_t)i * T_LEN, tin + i * LSTRIDE + lane);
        }
#endif
        cfence();
        wait_async0();   // fallback: input DMAs done; both paths: tout drained
        cfence();

        // -- sequential Holt-Winters scan over this tile
        #pragma unroll
        for (int tt = 0; tt < TILE; ++tt) {
            const float y = tin[lane * LSTRIDE + tt];
            float res;
            if (t0 == 0 && tt == 0) {
                trend = init_trend;
                prev  = y + trend;          // smooth0 = y ; res0 = smooth+trend
                res   = prev;
            } else {
                j = (j + 1 == SEASLEN) ? 0 : j + 1;
                const float season = season_s[j];
                const float d      = (y - season) - prev;
                const float s_new  = fmaf(alpha, d, prev);      // new smooth
                trend = fmaf(beta, s_new - prev, trend);        // new trend
                prev  = s_new + trend;
                res   = prev + season;
            }
            tout[tt * LSTRIDE + lane] = res;
        }
        cfence();
        wait_ds0();      // LDS result tile fully written before DMA reads it

        // -- drain output tile: instr i = series row (b0+i), lane = time column
        {
            float* gbase = out + (size_t)b0 * T_LEN + t0 + lane;
            #pragma unroll
            for (int i = 0; i < TILE; ++i)
                async_store_f32(gbase + (size_t)i * T_LEN, tout + lane * LSTRIDE + i);
        }
    }
    wait_async0();
}

// input order: 0 series[B*T] f32, 1 series_shifts[B] i32, 2 alpha f32,
//              3 beta f32, 4 gamma f32 (unused), 5 init_season[12] f32,
//              6 init_trend f32, 7 n_preds (unused, ==0)
extern "C" void kernel_launch(void* const* d_in, const int* in_sizes, int n_in,
                              void* d_out, int out_size, void* d_ws, size_t ws_size,
                              hipStream_t stream) {
    const float* series   = (const float*)d_in[0];
    const int*   shifts   = (const int*)  d_in[1];
    const float* alpha_p  = (const float*)d_in[2];
    const float* beta_p   = (const float*)d_in[3];
    const float* season_p = (const float*)d_in[5];
    const float* trend_p  = (const float*)d_in[6];
    float*       out      = (float*)d_out;

    const int B = in_sizes[1];          // 65536
    dim3 grid(B / BLOCK_T);             // 512 blocks of 128 threads (4 waves)
    dim3 block(BLOCK_T);
    holtwinters_scan<<<grid, block, 0, stream>>>(series, shifts, alpha_p, beta_p,
                                                 season_p, trend_p, out);
}